// SmoothL1LossWordAndSentence_9861244912209
// MI455X (gfx1250) — compile-verified
//
#include <hip/hip_runtime.h>

#define B_ 32
#define T_ 2048
#define D_ 512
#define TCHUNK 128   // 512 blocks x 8 waves = 4096 waves -> plenty of MLP for 23.3 TB/s

typedef __attribute__((ext_vector_type(2))) float v2f;
typedef __attribute__((ext_vector_type(8))) float v8f;

__device__ __forceinline__ float smooth_l1(float d) {
    float ad = __builtin_fabsf(d);
    return ad < 1.0f ? 0.5f * d * d : ad - 0.5f;
}

// ws layout (floats): [0] word-loss accumulator; [16, 16+B*D) psum; [16+B*D, 16+2*B*D) tsum
__global__ __launch_bounds__(256) void
smoothl1_partial_kernel(const float* __restrict__ preds,
                        const float* __restrict__ targs,
                        const int*   __restrict__ dlen,
                        float*       __restrict__ ws) {
    const int chunks = T_ / TCHUNK;
    const int b      = blockIdx.x / chunks;
    const int tbase  = (blockIdx.x % chunks) * TCHUNK;
    const int len    = dlen[b];                 // uniform across block

    __shared__ float s_word[8];
    const int wave = threadIdx.x >> 5;
    const int lane = threadIdx.x & 31;

    float* psum = ws + 16;
    float* tsum = ws + 16 + B_ * D_;
    float  wacc = 0.0f;

    if (tbase < len) {                          // block-uniform: no divergence
        int lim = tbase + TCHUNK;
        if (lim > len) lim = len;
        // largest tile-aligned end: tiles [tbase, tfull) are fully inside the mask
        const int tfull = tbase + ((lim - tbase) & ~3);

        const int n  = lane & 15;               // column of d-group (B-matrix N)
        const int kb = (lane >> 4) * 2;         // K rows {kb, kb+1} per f32 B layout
        const v2f ones = {1.0f, 1.0f};          // A = ones -> D rows = column sums of B

        for (int gi = 0; gi < 4; ++gi) {
            const int g = wave + gi * 8;        // d-group 0..31
            const int d = g * 16 + n;
            const float* pB = preds + (size_t)b * T_ * D_ + d;
            const float* tB = targs + (size_t)b * T_ * D_ + d;

            v8f cp = {0.f,0.f,0.f,0.f,0.f,0.f,0.f,0.f};
            v8f ct = {0.f,0.f,0.f,0.f,0.f,0.f,0.f,0.f};

            // ---- full tiles: unconditional, branch-free loads ----
#pragma unroll 2
            for (int t = tbase; t < tfull; t += 4) {
                const size_t r0 = (size_t)(t + kb) * D_;
                const size_t r1 = r0 + D_;
                float p0 = pB[r0];
                float p1 = pB[r1];
                float q0 = tB[r0];
                float q1 = tB[r1];

                v2f bp = {p0, p1};
                v2f bt = {q0, q1};
                cp = __builtin_amdgcn_wmma_f32_16x16x4_f32(false, ones, false, bp,
                                                           (short)0, cp, false, false);
                ct = __builtin_amdgcn_wmma_f32_16x16x4_f32(false, ones, false, bt,
                                                           (short)0, ct, false, false);
                wacc += smooth_l1(p0 - q0) + smooth_l1(p1 - q1);
            }

            // ---- at most one masked tail tile: clamped loads * {0,1} mask ----
            if (tfull < lim) {
                const int t0 = tfull + kb;
                const int t1 = t0 + 1;
                const float m0 = (t0 < len) ? 1.0f : 0.0f;
                const float m1 = (t1 < len) ? 1.0f : 0.0f;
                const int c0 = (t0 < len) ? t0 : (len - 1);   // in-bounds clamp
                const int c1 = (t1 < len) ? t1 : (len - 1);
                float p0 = pB[(size_t)c0 * D_] * m0;
                float p1 = pB[(size_t)c1 * D_] * m1;
                float q0 = tB[(size_t)c0 * D_] * m0;
                float q1 = tB[(size_t)c1 * D_] * m1;

                v2f bp = {p0, p1};
                v2f bt = {q0, q1};
                cp = __builtin_amdgcn_wmma_f32_16x16x4_f32(false, ones, false, bp,
                                                           (short)0, cp, false, false);
                ct = __builtin_amdgcn_wmma_f32_16x16x4_f32(false, ones, false, bt,
                                                           (short)0, ct, false, false);
                wacc += smooth_l1(p0 - q0) + smooth_l1(p1 - q1);
            }

            // A==ones => all 16 rows identical; c[0] holds the column-n sum on every lane
            // (row 0 on lanes 0-15, row 8 on lanes 16-31).
            float colp = cp[0];
            float colt = ct[0];
            if (lane < 16) atomicAdd(&psum[b * D_ + d], colp);
            else           atomicAdd(&tsum[b * D_ + d], colt);
        }
    }

    // block reduction of word-loss partials -> one atomic per block
    for (int off = 16; off > 0; off >>= 1) wacc += __shfl_xor(wacc, off, 32);
    if (lane == 0) s_word[wave] = wacc;
    __syncthreads();
    if (threadIdx.x == 0) {
        float tot = 0.0f;
        for (int i = 0; i < 8; ++i) tot += s_word[i];
        atomicAdd(&ws[0], tot);
    }
}

__global__ __launch_bounds__(1024) void
smoothl1_finalize_kernel(const int* __restrict__ dlen,
                         const float* __restrict__ ws,
                         float* __restrict__ out) {
    __shared__ float s[32];
    const float* psum = ws + 16;
    const float* tsum = ws + 16 + B_ * D_;

    float acc = 0.0f;
    for (int i = threadIdx.x; i < B_ * D_; i += 1024) {
        const int   bb  = i / D_;
        const float len = (float)dlen[bb];
        const float dm  = (psum[i] - tsum[i]) / len;   // pred_mean - targ_mean
        acc += smooth_l1(dm);
    }
    for (int off = 16; off > 0; off >>= 1) acc += __shfl_xor(acc, off, 32);
    const int wave = threadIdx.x >> 5;
    const int lane = threadIdx.x & 31;
    if (lane == 0) s[wave] = acc;
    __syncthreads();
    if (wave == 0) {
        float v = s[lane];
        for (int off = 16; off > 0; off >>= 1) v += __shfl_xor(v, off, 32);
        if (lane == 0) {
            long long sumlen = 0;
            for (int i = 0; i < B_; ++i) sumlen += dlen[i];
            const float word = ws[0] / ((float)sumlen * (float)D_);
            const float sent = v / ((float)B_ * (float)D_);   // (1/B) * sum_b mean_d
            out[0] = word + sent;
        }
    }
}

extern "C" void kernel_launch(void* const* d_in, const int* in_sizes, int n_in,
                              void* d_out, int out_size, void* d_ws, size_t ws_size,
                              hipStream_t stream) {
    (void)in_sizes; (void)n_in; (void)out_size; (void)ws_size;
    const float* preds = (const float*)d_in[0];
    const float* targs = (const float*)d_in[1];
    const int*   dlen  = (const int*)d_in[2];
    float*       out   = (float*)d_out;
    float*       ws    = (float*)d_ws;

    // zero accumulators every launch (graph-capturable, deterministic across replays)
    hipMemsetAsync(d_ws, 0, (size_t)(16 + 2 * B_ * D_) * sizeof(float), stream);

    dim3 grid(B_ * (T_ / TCHUNK));       // 512 blocks
    smoothl1_partial_kernel<<<grid, 256, 0, stream>>>(preds, targs, dlen, ws);
    smoothl1_finalize_kernel<<<1, 1024, 0, stream>>>(dlen, ws, out);
}